// CaptioningRNN_11407433138273
// MI455X (gfx1250) — compile-verified
//
#include <hip/hip_runtime.h>
#include <math.h>

typedef __attribute__((ext_vector_type(2))) float v2f;
typedef __attribute__((ext_vector_type(8))) float v8f;

constexpr int kN = 256;
constexpr int kT = 128;
constexpr int kD = 512;
constexpr int kH = 512;
constexpr int k4H = 2048;

// ---------------------------------------------------------------------------
// init: h0 = c0 = mean over the 16 spatial cells of A_flat (N,H,16)
// ---------------------------------------------------------------------------
__global__ __launch_bounds__(256) void init_h0_kernel(
    const float* __restrict__ A, float* __restrict__ h, float* __restrict__ c) {
  int idx = blockIdx.x * blockDim.x + threadIdx.x;  // over N*H
  if (idx >= kN * kH) return;
  const float* a = A + (size_t)idx * 16;
  float s = 0.f;
#pragma unroll
  for (int p = 0; p < 16; ++p) s += a[p];
  s *= (1.0f / 16.0f);
  h[idx] = s;
  c[idx] = s;
}

// ---------------------------------------------------------------------------
// attention: per n, scores[p] = sum_h h[n,h]*A[n,h,p] / sqrt(H); softmax over
// 16 cells; attn[n,h] = sum_p A[n,h,p] * w[p].  One block per n.
// ---------------------------------------------------------------------------
__global__ __launch_bounds__(256) void attention_kernel(
    const float* __restrict__ A, const float* __restrict__ h,
    float* __restrict__ attn) {
  const int n = blockIdx.x;
  const int tid = threadIdx.x;
  const int lane = tid & 31;
  const float* An = A + (size_t)n * kH * 16;
  const float* hn = h + (size_t)n * kH;

  __shared__ float s_scores[16];
  __shared__ float s_w[16];
  if (tid < 16) s_scores[tid] = 0.f;
  __syncthreads();

  float part[16];
#pragma unroll
  for (int p = 0; p < 16; ++p) part[p] = 0.f;
  for (int hh = tid; hh < kH; hh += blockDim.x) {
    float hv = hn[hh];
    const float* row = An + (size_t)hh * 16;
#pragma unroll
    for (int p = 0; p < 16; ++p) part[p] = fmaf(hv, row[p], part[p]);
  }
  // wave32 butterfly reduction, then one LDS atomic per wave per p
#pragma unroll
  for (int p = 0; p < 16; ++p) {
    float v = part[p];
#pragma unroll
    for (int off = 16; off > 0; off >>= 1) v += __shfl_xor(v, off, 32);
    if (lane == 0) atomicAdd(&s_scores[p], v);
  }
  __syncthreads();

  if (tid == 0) {
    const float scale = 0.04419417382415922f;  // 1/sqrt(512)
    float sc[16];
    float m = -INFINITY;
#pragma unroll
    for (int p = 0; p < 16; ++p) {
      sc[p] = s_scores[p] * scale;
      m = fmaxf(m, sc[p]);
    }
    float sum = 0.f;
#pragma unroll
    for (int p = 0; p < 16; ++p) {
      sc[p] = __expf(sc[p] - m);
      sum += sc[p];
    }
    float inv = 1.0f / sum;
#pragma unroll
    for (int p = 0; p < 16; ++p) s_w[p] = sc[p] * inv;
  }
  __syncthreads();

  for (int hh = tid; hh < kH; hh += blockDim.x) {
    const float* row = An + (size_t)hh * 16;
    float acc = 0.f;
#pragma unroll
    for (int p = 0; p < 16; ++p) acc = fmaf(row[p], s_w[p], acc);
    attn[(size_t)n * kH + hh] = acc;
  }
}

// ---------------------------------------------------------------------------
// one LSTM step: grid (N/16, H/16), 256 threads = 8 waves.
// wave w: gate g = w&3 (i,f,o,g), K-half g2 = w>>2.  Each wave accumulates a
// 16x16 fp32 tile over 768 K-values (half of D + H + H) as 192 WMMAs.
// A operand (x_t / h / attn rows) is staged in LDS chunks of 16x64 via
// GLOBAL_LOAD_ASYNC_TO_LDS_B128 (double-buffered, ASYNCcnt), shared by the 4
// gate waves of each K-half.  B (weights) streams from L2.
// ---------------------------------------------------------------------------
__global__ __launch_bounds__(256) void step_kernel(
    const float* __restrict__ x,      // (N,T,D)
    const float* __restrict__ Wx,     // (D,4H)
    const float* __restrict__ Wh,     // (H,4H)
    const float* __restrict__ Wattn,  // (H,4H)
    const float* __restrict__ bias,   // (4H)
    const float* __restrict__ h_in,   // (N,H)
    const float* __restrict__ attn,   // (N,H)
    float* __restrict__ h_out,        // (N,H)
    float* __restrict__ c,            // (N,H) in/out
    float* __restrict__ hs,           // (N,T,H) output
    int t) {
  const int tid = threadIdx.x;
  const int wave = tid >> 5;
  const int lane = tid & 31;
  const int gate = wave & 3;  // i,f,o,g
  const int g2 = wave >> 2;   // K-half
  const int col = lane & 15;  // A row index == B/D column index
  const int hi = lane >> 4;
  const int n0 = blockIdx.x * 16;
  const int hTile = blockIdx.y;
  const int jn = gate * kH + hTile * 16 + col;  // column in 4H space

  // A-chunk staging buffers: [double-buffer][K-half][row][64 + 4 pad]
  // pitch 68 floats = 272B -> 4-bank stagger per row, conflict-free b64 reads.
  __shared__ __align__(16) float abuf[2][2][16][68];
  __shared__ float gbuf[8][16][17];

  const float* srcb[3];
  int ldsrc[3];
  srcb[0] = x + (size_t)n0 * kT * kD + (size_t)t * kD;
  ldsrc[0] = kT * kD;
  srcb[1] = h_in + (size_t)n0 * kH;
  ldsrc[1] = kH;
  srcb[2] = attn + (size_t)n0 * kH;
  ldsrc[2] = kH;
  const float* wseg[3] = {Wx, Wh, Wattn};

  const int tid2 = tid & 127;  // thread id within K-half group (4 waves)

  // stage one 16x64 chunk (seg, koff) for this K-half group into abuf[buf]
  auto stage = [&](int buf, int cidx) {
    const int seg = cidx >> 2;                  // 0:x 1:h 2:attn
    const int koff = (cidx & 3) * 64 + g2 * 256;  // K offset within segment
    const float* sb = srcb[seg];
    const int ld = ldsrc[seg];
#pragma unroll
    for (int j = 0; j < 2; ++j) {
      const int s = tid2 + j * 128;  // 256 float4 slots
      const int row = s >> 4;
      const int q = s & 15;
      const float* gp = sb + (size_t)row * ld + koff + q * 4;
      unsigned int la = (unsigned int)(size_t)&abuf[buf][g2][row][q * 4];
      unsigned long long ga = (unsigned long long)(size_t)gp;
      asm volatile("global_load_async_to_lds_b128 %0, %1, off" ::"v"(la),
                   "v"(ga)
                   : "memory");
    }
  };

  v8f acc = {0.f, 0.f, 0.f, 0.f, 0.f, 0.f, 0.f, 0.f};

  stage(0, 0);
  asm volatile("s_wait_asynccnt 0" ::: "memory");
  __syncthreads();

  for (int cidx = 0; cidx < 12; ++cidx) {
    const int buf = cidx & 1;
    if (cidx + 1 < 12) stage(buf ^ 1, cidx + 1);

    const int seg = cidx >> 2;
    const int koff = (cidx & 3) * 64 + g2 * 256;
    const float* wb = wseg[seg] + (size_t)(koff + 2 * hi) * k4H + jn;
    const float* ab = &abuf[buf][g2][col][2 * hi];
#pragma unroll
    for (int kk = 0; kk < 64; kk += 4) {
      v2f a = *(const v2f*)(ab + kk);  // ds_load_b64
      v2f b;
      b.x = wb[(size_t)kk * k4H];
      b.y = wb[(size_t)(kk + 1) * k4H];
      acc = __builtin_amdgcn_wmma_f32_16x16x4_f32(
          /*neg_a=*/false, a, /*neg_b=*/false, b,
          /*c_mod=*/(short)0, acc, /*reuse_a=*/false, /*reuse_b=*/false);
    }

    asm volatile("s_wait_asynccnt 0" ::: "memory");
    __syncthreads();
  }

  // park each wave's tile; C/D layout: vgpr r -> row r + 8*hi, col = lane&15
#pragma unroll
  for (int r = 0; r < 8; ++r) gbuf[wave][r + 8 * hi][col] = acc[r];
  __syncthreads();

  // 256 threads, one element each: reduce K-halves, gate math, advance state
  {
    const int rr = tid >> 4;  // n offset within tile
    const int cc = tid & 15;  // h offset within tile
    const int n = n0 + rr;
    const int hcol = hTile * 16 + cc;
    const float ai = gbuf[0][rr][cc] + gbuf[4][rr][cc] + bias[0 * kH + hcol];
    const float af = gbuf[1][rr][cc] + gbuf[5][rr][cc] + bias[1 * kH + hcol];
    const float ao = gbuf[2][rr][cc] + gbuf[6][rr][cc] + bias[2 * kH + hcol];
    const float ag = gbuf[3][rr][cc] + gbuf[7][rr][cc] + bias[3 * kH + hcol];
    const float ig = 1.0f / (1.0f + __expf(-ai));
    const float fg = 1.0f / (1.0f + __expf(-af));
    const float og = 1.0f / (1.0f + __expf(-ao));
    const float gg = tanhf(ag);
    const size_t ci = (size_t)n * kH + hcol;
    const float cn = fg * c[ci] + ig * gg;
    c[ci] = cn;
    const float hn = og * tanhf(cn);
    h_out[ci] = hn;
    hs[((size_t)n * kT + t) * kH + hcol] = hn;
  }
}

// ---------------------------------------------------------------------------
extern "C" void kernel_launch(void* const* d_in, const int* in_sizes, int n_in,
                              void* d_out, int out_size, void* d_ws,
                              size_t ws_size, hipStream_t stream) {
  const float* x = (const float*)d_in[0];      // (N,T,D)
  const float* A = (const float*)d_in[1];      // (N,H,4,4)
  const float* Wx = (const float*)d_in[2];     // (D,4H)
  const float* Wh = (const float*)d_in[3];     // (H,4H)
  const float* Wattn = (const float*)d_in[4];  // (H,4H)
  const float* b = (const float*)d_in[5];      // (4H)
  float* out = (float*)d_out;                  // (N,T,H)

  float* hbuf0 = (float*)d_ws;             // N*H
  float* hbuf1 = hbuf0 + (size_t)kN * kH;  // N*H
  float* cbuf = hbuf1 + (size_t)kN * kH;   // N*H
  float* attnbuf = cbuf + (size_t)kN * kH;

  init_h0_kernel<<<(kN * kH + 255) / 256, 256, 0, stream>>>(A, hbuf0, cbuf);

  dim3 gstep(kN / 16, kH / 16);
  for (int t = 0; t < kT; ++t) {
    const float* hin = (t & 1) ? hbuf1 : hbuf0;
    float* hout = (t & 1) ? hbuf0 : hbuf1;
    attention_kernel<<<kN, 256, 0, stream>>>(A, hin, attnbuf);
    step_kernel<<<gstep, 256, 0, stream>>>(x, Wx, Wh, Wattn, b, hin, attnbuf,
                                           hout, cbuf, out, t);
  }
}